// GNNModel_62663572849123
// MI455X (gfx1250) — compile-verified
//
#include <hip/hip_runtime.h>
#include <hip/hip_bf16.h>
#include <math.h>

typedef __attribute__((ext_vector_type(2))) float v2f;
typedef __attribute__((ext_vector_type(8))) float v8f;

#define NNODES 100000
#define DFEAT  64
#define NGRAPH 512
#define DOUT   10
#define BNEPS  1e-5f

static __device__ __forceinline__ void atomAddF(float* p, float v) {
  __hip_atomic_fetch_add(p, v, __ATOMIC_RELAXED, __HIP_MEMORY_SCOPE_AGENT);
}

// ---------------------------------------------------------------- utilities
__global__ void k_zero(float* __restrict__ p, int n) {
  int t = blockIdx.x * blockDim.x + threadIdx.x;
  if (t < n) p[t] = 0.0f;
}

__global__ void k_fill1(float* __restrict__ p, int n) {
  int t = blockIdx.x * blockDim.x + threadIdx.x;
  if (t < n) p[t] = 1.0f;   // self-loop contributes 1 to every degree
}

// deg[dst] += 1 for each real edge
__global__ void k_deg(const long long* __restrict__ dst, float* __restrict__ deg, int E) {
  int t = blockIdx.x * blockDim.x + threadIdx.x;
  if (t < E) atomAddF(&deg[(int)dst[t]], 1.0f);
}

// deg -> dinv (in place)
__global__ void k_dinv(float* __restrict__ d, int n) {
  int t = blockIdx.x * blockDim.x + threadIdx.x;
  if (t < n) { float v = d[t]; d[t] = (v > 0.0f) ? rsqrtf(v) : 0.0f; }
}

// ------------------------------------------------- WMMA GEMM: H = act(A)@W
// act(A) = folded-BN affine + ReLU, applied ONCE per element while staging the
// 32x64 A tile into LDS with coalesced b128 loads. Inner loop is pure
// ds_load + V_WMMA_F32_16X16X4_F32 (16 steps of K=4).
// Block: 256 threads = 8 waves; wave (tile_m,tile_n) owns a 16x16 output tile.
template <int USE_AFFINE>
__global__ void __launch_bounds__(256)
k_gemm_wmma(const float* __restrict__ A, const float* __restrict__ W,
            float* __restrict__ H, const float* __restrict__ scale,
            const float* __restrict__ shift) {
  __shared__ float ldsW[DFEAT * DFEAT];   // 16 KB: whole weight matrix
  __shared__ float ldsA[32 * DFEAT];      //  8 KB: activated A tile

  // stage W: 4096 floats = 1024 float4, 4 per thread
  {
    const float4* __restrict__ W4 = (const float4*)W;
    float4* ldsW4 = (float4*)ldsW;
    #pragma unroll
    for (int i = 0; i < 4; ++i)
      ldsW4[threadIdx.x + i * 256] = W4[threadIdx.x + i * 256];
  }
  // stage A tile: 2048 floats = 512 float4, 2 per thread; fuse BN+ReLU
  {
    const float4* __restrict__ A4 =
        (const float4*)(A + (size_t)blockIdx.x * 32 * DFEAT);
    float4* ldsA4 = (float4*)ldsA;
    #pragma unroll
    for (int j = 0; j < 2; ++j) {
      const int i = threadIdx.x + j * 256;
      float4 v = A4[i];
      if (USE_AFFINE) {
        const int c = (i * 4) & 63;
        v.x = fmaxf(v.x * scale[c]     + shift[c],     0.0f);
        v.y = fmaxf(v.y * scale[c + 1] + shift[c + 1], 0.0f);
        v.z = fmaxf(v.z * scale[c + 2] + shift[c + 2], 0.0f);
        v.w = fmaxf(v.w * scale[c + 3] + shift[c + 3], 0.0f);
      }
      ldsA4[i] = v;
    }
  }
  __syncthreads();

  const int lane   = threadIdx.x & 31;
  const int wave   = threadIdx.x >> 5;
  const int tile_m = wave >> 2;              // 0..1
  const int tile_n = wave & 3;               // 0..3
  const int khalf  = (lane >> 4) * 2;        // lanes 16-31 hold K+2,K+3
  const int mloc   = tile_m * 16 + (lane & 15);
  const int ncol   = tile_n * 16 + (lane & 15);

  v8f acc = {};
  #pragma unroll
  for (int k0 = 0; k0 < DFEAT; k0 += 4) {
    const int ka = k0 + khalf;
    v2f a = { ldsA[mloc * DFEAT + ka], ldsA[mloc * DFEAT + ka + 1] };
    v2f b = { ldsW[ka * DFEAT + ncol], ldsW[(ka + 1) * DFEAT + ncol] };
    acc = __builtin_amdgcn_wmma_f32_16x16x4_f32(
        /*neg_a=*/false, a, /*neg_b=*/false, b,
        /*c_mod=*/(short)0, acc, /*reuse_a=*/false, /*reuse_b=*/false);
  }

  // C/D layout: VGPR i -> M=i (lanes 0-15), M=i+8 (lanes 16-31); N = lane&15
  const int mb = blockIdx.x * 32 + tile_m * 16 + (lane >> 4) * 8;
  #pragma unroll
  for (int i = 0; i < 8; ++i)
    H[(size_t)(mb + i) * DFEAT + ncol] = acc[i];
}

// ---------------------------------------------- message passing (scatter)
// self-loop init: agg[i] = h[i] * dinv[i]^2
__global__ void k_selfinit(const float* __restrict__ h, const float* __restrict__ dinv,
                           float* __restrict__ agg, int n64) {
  int t = blockIdx.x * blockDim.x + threadIdx.x;
  if (t < n64) {
    int i = t >> 6;
    float dv = dinv[i];
    agg[t] = h[t] * dv * dv;
  }
}

// one wave per edge; each lane handles 2 of the 64 channels
__global__ void __launch_bounds__(256)
k_edge_scatter(const float* __restrict__ h, const float* __restrict__ dinv,
               const long long* __restrict__ src, const long long* __restrict__ dst,
               float* __restrict__ agg, int E) {
  int t = blockIdx.x * 256 + threadIdx.x;
  int e = t >> 5;
  int lane = t & 31;
  if (e >= E) return;
  int s = (int)src[e];
  int d = (int)dst[e];
  float nrm = dinv[s] * dinv[d];
  const float2 hv = *(const float2*)(h + (size_t)s * DFEAT + lane * 2);
  float* out = agg + (size_t)d * DFEAT + lane * 2;
  atomAddF(out,     hv.x * nrm);
  atomAddF(out + 1, hv.y * nrm);
}

// ------------------------------------------------------ BatchNorm statistics
__global__ void __launch_bounds__(256)
k_bn_stats(const float* __restrict__ x, float* __restrict__ sums,
           float* __restrict__ sumsq, int nrows) {
  __shared__ float s_s[256];
  __shared__ float s_q[256];
  const int c = threadIdx.x & 63;
  const int rg = threadIdx.x >> 6;       // 0..3
  float a = 0.0f, b = 0.0f;
  for (int r = blockIdx.x * 4 + rg; r < nrows; r += gridDim.x * 4) {
    float v = x[(size_t)r * DFEAT + c];
    a += v;
    b += v * v;
  }
  s_s[threadIdx.x] = a;
  s_q[threadIdx.x] = b;
  __syncthreads();
  if (threadIdx.x < 64) {
    a = s_s[c] + s_s[64 + c] + s_s[128 + c] + s_s[192 + c];
    b = s_q[c] + s_q[64 + c] + s_q[128 + c] + s_q[192 + c];
    atomAddF(&sums[c], a);
    atomAddF(&sumsq[c], b);
  }
}

// scale = gamma * rsqrt(var+eps); shift = beta - mu*scale
__global__ void k_bn_finalize(const float* __restrict__ sums, const float* __restrict__ sumsq,
                              const float* __restrict__ gamma, const float* __restrict__ beta,
                              float* __restrict__ scale, float* __restrict__ shift, float invN) {
  int c = threadIdx.x;
  if (c < DFEAT) {
    float mu  = sums[c] * invN;
    float var = sumsq[c] * invN - mu * mu;
    float sc  = gamma[c] * rsqrtf(var + BNEPS);
    scale[c] = sc;
    shift[c] = beta[c] - mu * sc;
  }
}

// ----------------------------------------------------- pooling (BN3 fused)
__global__ void k_pool(const float* __restrict__ agg, const float* __restrict__ scale,
                       const float* __restrict__ shift, const long long* __restrict__ batch,
                       float* __restrict__ pooled, float* __restrict__ cnt, int n64) {
  int t = blockIdx.x * blockDim.x + threadIdx.x;
  if (t < n64) {
    int i = t >> 6;
    int c = t & 63;
    int g = (int)batch[i];
    float v = fmaxf(agg[t] * scale[c] + shift[c], 0.0f);
    atomAddF(&pooled[g * DFEAT + c], v);
    if (c == 0) atomAddF(&cnt[g], 1.0f);
  }
}

// ------------------------------------------------------------------ FC head
__global__ void k_fc(const float* __restrict__ pooled, const float* __restrict__ cnt,
                     const float* __restrict__ Wfc, const float* __restrict__ bfc,
                     float* __restrict__ out) {
  int t = blockIdx.x * blockDim.x + threadIdx.x;
  if (t < NGRAPH * DOUT) {
    int g = t / DOUT;
    int o = t % DOUT;
    float inv = 1.0f / fmaxf(cnt[g], 1.0f);
    float s = bfc[o];
    #pragma unroll
    for (int k = 0; k < DFEAT; ++k)
      s += pooled[g * DFEAT + k] * inv * Wfc[k * DOUT + o];
    out[t] = s;
  }
}

// ================================================================== launch
extern "C" void kernel_launch(void* const* d_in, const int* in_sizes, int n_in,
                              void* d_out, int out_size, void* d_ws, size_t ws_size,
                              hipStream_t stream) {
  const float*      x    = (const float*)d_in[0];
  const long long*  ei   = (const long long*)d_in[1];   // [2, E]
  const long long*  batch= (const long long*)d_in[2];
  const float*      W1   = (const float*)d_in[3];
  const float*      g1   = (const float*)d_in[5];
  const float*      be1  = (const float*)d_in[6];
  const float*      W2   = (const float*)d_in[7];
  const float*      g2   = (const float*)d_in[9];
  const float*      be2  = (const float*)d_in[10];
  const float*      W3   = (const float*)d_in[11];
  const float*      g3   = (const float*)d_in[13];
  const float*      be3  = (const float*)d_in[14];
  const float*      Wfc  = (const float*)d_in[15];
  const float*      bfc  = (const float*)d_in[16];
  float*            out  = (float*)d_out;

  const int N  = NNODES;
  const int E  = in_sizes[1] / 2;
  const long long* src = ei;
  const long long* dst = ei + E;

  // workspace layout (floats)
  float* ws    = (float*)d_ws;
  const size_t N64 = (size_t)N * DFEAT;
  float* dinv  = ws;                              // N (degree, then rsqrt)
  float* hbuf  = ws + ((N + 255) / 256) * 256;    // N*64
  float* agg   = hbuf + N64;                      // N*64
  float* sums  = agg + N64;                       // 64
  float* sumsq = sums + DFEAT;                    // 64
  float* scale = sumsq + DFEAT;                   // 64
  float* shift = scale + DFEAT;                   // 64
  float* pooled= shift + DFEAT;                   // G*64
  float* cnt   = pooled + NGRAPH * DFEAT;         // G

  const int B = 256;
  const int gN    = (N + B - 1) / B;
  const int gN64  = (int)((N64 + B - 1) / B);
  const int gE    = (E + B - 1) / B;
  const int gE32  = (int)(((size_t)E * 32 + B - 1) / B);
  const int gGemm = N / 32;                       // 100000/32 = 3125 exact
  const float invN = 1.0f / (float)N;

  // ---- degree / normalization prep
  k_fill1<<<gN, B, 0, stream>>>(dinv, N);
  k_deg<<<gE, B, 0, stream>>>(dst, dinv, E);
  k_dinv<<<gN, B, 0, stream>>>(dinv, N);

  // ---- layer 1
  k_gemm_wmma<0><<<gGemm, B, 0, stream>>>(x, W1, hbuf, scale, shift);
  k_selfinit<<<gN64, B, 0, stream>>>(hbuf, dinv, agg, (int)N64);
  k_edge_scatter<<<gE32, B, 0, stream>>>(hbuf, dinv, src, dst, agg, E);
  k_zero<<<1, 128, 0, stream>>>(sums, 128);
  k_bn_stats<<<512, B, 0, stream>>>(agg, sums, sumsq, N);
  k_bn_finalize<<<1, 64, 0, stream>>>(sums, sumsq, g1, be1, scale, shift, invN);

  // ---- layer 2 (BN1+ReLU fused into GEMM A-stage)
  k_gemm_wmma<1><<<gGemm, B, 0, stream>>>(agg, W2, hbuf, scale, shift);
  k_selfinit<<<gN64, B, 0, stream>>>(hbuf, dinv, agg, (int)N64);
  k_edge_scatter<<<gE32, B, 0, stream>>>(hbuf, dinv, src, dst, agg, E);
  k_zero<<<1, 128, 0, stream>>>(sums, 128);
  k_bn_stats<<<512, B, 0, stream>>>(agg, sums, sumsq, N);
  k_bn_finalize<<<1, 64, 0, stream>>>(sums, sumsq, g2, be2, scale, shift, invN);

  // ---- layer 3
  k_gemm_wmma<1><<<gGemm, B, 0, stream>>>(agg, W3, hbuf, scale, shift);
  k_selfinit<<<gN64, B, 0, stream>>>(hbuf, dinv, agg, (int)N64);
  k_edge_scatter<<<gE32, B, 0, stream>>>(hbuf, dinv, src, dst, agg, E);
  k_zero<<<1, 128, 0, stream>>>(sums, 128);
  k_bn_stats<<<512, B, 0, stream>>>(agg, sums, sumsq, N);
  k_bn_finalize<<<1, 64, 0, stream>>>(sums, sumsq, g3, be3, scale, shift, invN);

  // ---- pool (BN3+ReLU fused) + FC
  k_zero<<<(NGRAPH * DFEAT + NGRAPH + B - 1) / B, B, 0, stream>>>(pooled, NGRAPH * DFEAT + NGRAPH);
  k_pool<<<gN64, B, 0, stream>>>(agg, scale, shift, batch, pooled, cnt, (int)N64);
  k_fc<<<(NGRAPH * DOUT + B - 1) / B, B, 0, stream>>>(pooled, cnt, Wfc, bfc, out);
}